// HoughNet_Head_13941463843331
// MI455X (gfx1250) — compile-verified
//
#include <hip/hip_runtime.h>
#include <hip/hip_bf16.h>
#include <math.h>

// ---------------- problem constants ----------------
#define C_IN 256
#define Hh   256
#define Ww   512
#define HW   (Hh * Ww)          // 131072 pixels per image
#define NPIX (2 * HW)           // 262144 pixels total (B=2)

typedef __attribute__((ext_vector_type(2))) float v2f;
typedef __attribute__((ext_vector_type(8))) float v8f;

// ---------------- compile-time Hough vote-field table ----------------
// Region of a filter cell at offset (ys, xs) from center (ANGLE=90 -> quadrant
// bins via sign tests; rings at r^2 <= {4,64,256,1024}; grid max r^2 = 512).
constexpr int regionOf(int ys, int xs) {
    int r2 = xs * xs + ys * ys;
    if (r2 <= 4) return 0;
    int bin = (ys > 0) ? ((xs > 0) ? 0 : 1)
            : (ys < 0) ? ((xs < 0) ? 2 : 3)
                       : ((xs > 0) ? 0 : 2);   // ys==0 row: theta 0 or 180
    int ring = (r2 <= 64) ? 0 : ((r2 <= 256) ? 1 : 2);
    return 1 + ring * 4 + bin;
}

struct HTab {
    short dy[1089];     // tap offsets (already includes the kernel flip:
    short dx[1089];     //  tap = (-ys, -xs) for filter cell (ys, xs))
    int   start[14];    // taps grouped by source channel r
    float w[13];        // 1 / count(region)
};

constexpr HTab makeHTab() {
    HTab t{};
    int cnt[13] = {};
    for (int u = 0; u < 33; ++u)
        for (int v = 0; v < 33; ++v)
            cnt[regionOf(u - 16, v - 16)]++;
    t.start[0] = 0;
    for (int r = 0; r < 13; ++r) t.start[r + 1] = t.start[r] + cnt[r];
    int pos[13] = {};
    for (int r = 0; r < 13; ++r) pos[r] = t.start[r];
    for (int u = 0; u < 33; ++u)
        for (int v = 0; v < 33; ++v) {
            int r = regionOf(u - 16, v - 16);
            t.dy[pos[r]] = (short)(16 - u);   // SAME-conv with flipped kernel
            t.dx[pos[r]] = (short)(16 - v);
            pos[r]++;
        }
    for (int r = 0; r < 13; ++r) t.w[r] = 1.0f / (float)cnt[r];
    return t;
}

__constant__ HTab c_ht = makeHTab();

// ---------------- Kernel 1: fused 1x1 convs via f32 WMMA ----------------
// GEMM: D[16 ch x 16 px] += A[16 x 4] * B[4 x 16], K = 256 in chunks of 4.
// One wave handles 64 consecutive pixels (4 N-tiles), reusing each A fragment
// 4x. ch 0..12 -> ReLU -> evident (workspace); ch 13 -> x_reg; ch 14,15 -> x_off.
__global__ __launch_bounds__(256) void head_gemm_wmma(
    const float* __restrict__ x,
    const float* __restrict__ pos_w, const float* __restrict__ pos_b,
    const float* __restrict__ reg_w, const float* __restrict__ reg_b,
    const float* __restrict__ off_w, const float* __restrict__ off_b,
    float* __restrict__ out, float* __restrict__ evident)
{
    __shared__ float W16[16 * 257];   // stride 257: conflict-free column reads
    __shared__ float bias[16];

    const int tid = threadIdx.x;

    // Stack [pos_w; reg_w; off_w] -> W16[16][256]
    for (int i = tid; i < 16 * 256; i += 256) {
        int m = i >> 8, k = i & 255;
        float wv;
        if (m < 13)       wv = pos_w[m * 256 + k];
        else if (m == 13) wv = reg_w[k];
        else              wv = off_w[(m - 14) * 256 + k];
        W16[m * 257 + k] = wv;
    }
    if (tid < 16)
        bias[tid] = (tid < 13) ? pos_b[tid] : (tid == 13 ? reg_b[0] : off_b[tid - 14]);
    __syncthreads();

    const int gr   = blockIdx.x;        // 0..511 global image row
    const int b    = gr >> 8;           // batch
    const int y    = gr & 255;
    const int wave = tid >> 5;
    const int lane = tid & 31;
    const int m    = lane & 15;         // A row (output channel 0..15 half)
    const int hi   = lane >> 4;         // lane half
    const int n    = lane & 15;         // B/D column (pixel within tile)

    const float* xb  = x + (size_t)b * C_IN * HW;
    const int    px0 = y * Ww + wave * 64;     // 64 pixels per wave

    v8f acc[4] = {v8f{}, v8f{}, v8f{}, v8f{}};

    #pragma unroll 4
    for (int kc = 0; kc < 64; ++kc) {
        const int k0 = kc * 4;
        // A fragment (16x4): VGPR0 holds K={0,2} across lane halves, VGPR1 K={1,3}
        v2f a;
        a.x = W16[m * 257 + k0 + 2 * hi];
        a.y = W16[m * 257 + k0 + 2 * hi + 1];

        // B fragments (4x16): lane halves split K; .x/.y are consecutive K rows
        const float* xp = xb + (size_t)(k0 + 2 * hi) * HW + px0 + n;
        __builtin_prefetch(xp + 4 * (size_t)HW, 0, 0);   // next K stripe
        v2f bf[4];
        #pragma unroll
        for (int t = 0; t < 4; ++t) {
            bf[t].x = xp[t * 16];
            bf[t].y = xp[(size_t)HW + t * 16];
        }
        #pragma unroll
        for (int t = 0; t < 4; ++t)
            acc[t] = __builtin_amdgcn_wmma_f32_16x16x4_f32(
                false, a, false, bf[t], (short)0, acc[t], false, false);
    }

    // Epilogue: D VGPR v, lanes 0-15 -> channel v, lanes 16-31 -> channel v+8.
    float*       ev        = evident + (size_t)b * 13 * HW;
    const size_t obase_reg = (size_t)NPIX + (size_t)b * HW;                    // x_reg
    const size_t obase_off = (size_t)NPIX + (size_t)NPIX + (size_t)b * 2 * HW; // x_off

    #pragma unroll
    for (int t = 0; t < 4; ++t) {
        const int p = px0 + t * 16 + n;
        #pragma unroll
        for (int v = 0; v < 8; ++v) {
            const int ch  = v + 8 * hi;
            const float val = acc[t][v] + bias[ch];
            if (ch < 13)
                ev[(size_t)ch * HW + p] = fmaxf(val, 0.0f);        // ReLU -> evident
            else if (ch == 13)
                out[obase_reg + p] = val;                          // x_reg
            else
                out[obase_off + (size_t)(ch - 14) * HW + p] = val; // x_off
        }
    }
}

// ---------------- Kernel 2: Hough vote (33x33, channel-select) + sigmoid ----
// 32x32 output tile per block; per source channel r, stage a 64x64 halo slab
// in LDS, then accumulate only region-r taps (grouped constexpr table).
// Each thread owns 4 CONSECUTIVE rows so its 4 tap reads are 64 floats apart
// (256 B) -> hardware DS_LOAD_2ADDR_B32 pairs (8-bit offset indices 0/64 and
// 128/192), halving DS instruction count vs. one ds_load_b32 per read.
__global__ __launch_bounds__(256) void hough_vote_sigmoid(
    const float* __restrict__ ev, float* __restrict__ out)
{
    __shared__ float slab[64 * 64];

    const int tid = threadIdx.x;
    const int lx  = tid & 31;       // tile x
    const int lyq = tid >> 5;       // 0..7; thread owns rows lyq*4 + {0,1,2,3}
    const int gx0 = blockIdx.x * 32;
    const int gr0 = blockIdx.y * 32;   // global row tile over B*H
    const int b   = gr0 >> 8;          // 256 % 32 == 0: tiles never straddle batch
    const int y0  = gr0 & 255;

    const float* evb = ev + (size_t)b * 13 * HW;

    float acc[4] = {0.f, 0.f, 0.f, 0.f};

    for (int r = 0; r < 13; ++r) {
        __syncthreads();
        const float* evc = evb + (size_t)r * HW;
        #pragma unroll
        for (int i = 0; i < 16; ++i) {           // 4096 elems / 256 threads
            const int idx = tid + i * 256;
            const int sy = idx >> 6, sx = idx & 63;
            const int gy = y0 - 16 + sy;
            const int gx = gx0 - 16 + sx;
            float v = 0.0f;
            if (gy >= 0 && gy < Hh && gx >= 0 && gx < Ww)
                v = evc[gy * Ww + gx];
            slab[idx] = v;
        }
        __syncthreads();

        const float wr = c_ht.w[r];
        const int t0 = c_ht.start[r], t1 = c_ht.start[r + 1];
        float s[4] = {0.f, 0.f, 0.f, 0.f};
        #pragma unroll 2
        for (int t = t0; t < t1; ++t) {          // uniform (scalar) tap offsets
            const int dy = c_ht.dy[t];
            const int dx = c_ht.dx[t];
            const int base = (lyq * 4 + 16 + dy) * 64 + (lx + 16 + dx);
            s[0] += slab[base];                  // -> ds_load_2addr pairs
            s[1] += slab[base + 64];
            s[2] += slab[base + 128];
            s[3] += slab[base + 192];
        }
        #pragma unroll
        for (int j = 0; j < 4; ++j) acc[j] += wr * s[j];
    }

    #pragma unroll
    for (int j = 0; j < 4; ++j) {
        const int oy = y0 + lyq * 4 + j;
        const float sg = 1.0f / (1.0f + __expf(-acc[j]));
        out[(size_t)b * HW + (size_t)oy * Ww + gx0 + lx] = sg;
    }
}

// ---------------- launch ----------------
extern "C" void kernel_launch(void* const* d_in, const int* in_sizes, int n_in,
                              void* d_out, int out_size, void* d_ws, size_t ws_size,
                              hipStream_t stream) {
    const float* x     = (const float*)d_in[0];
    const float* pos_w = (const float*)d_in[1];
    const float* pos_b = (const float*)d_in[2];
    const float* reg_w = (const float*)d_in[3];
    const float* reg_b = (const float*)d_in[4];
    const float* off_w = (const float*)d_in[5];
    const float* off_b = (const float*)d_in[6];
    float* out = (float*)d_out;
    float* ev  = (float*)d_ws;   // evident [2,13,256,512] f32 = 13.6 MB

    head_gemm_wmma<<<512, 256, 0, stream>>>(x, pos_w, pos_b, reg_w, reg_b,
                                            off_w, off_b, out, ev);
    hough_vote_sigmoid<<<dim3(Ww / 32, (2 * Hh) / 32), 256, 0, stream>>>(ev, out);
}